// GCN_38491496907229
// MI455X (gfx1250) — compile-verified
//
#include <hip/hip_runtime.h>

typedef __attribute__((ext_vector_type(2))) float v2f;
typedef __attribute__((ext_vector_type(4))) float v4f;
typedef __attribute__((ext_vector_type(8))) float v8f;

static constexpr int NNODES  = 100000;
static constexpr int NEDGES  = 3200000;
static constexpr int NGRAPHS = 1024;
static constexpr int DIN     = 128;
static constexpr int DHID    = 64;
static constexpr int NCLS    = 10;

// ---------------- degree / normalization (shared across layers) ----------------
__global__ void k_deg_init(float* __restrict__ deg) {
  int i = blockIdx.x * blockDim.x + threadIdx.x;
  if (i < NNODES) deg[i] = 1.0f;  // self loop
}

__global__ void k_deg_edges(const long long* __restrict__ dst, float* __restrict__ deg) {
  int i = blockIdx.x * blockDim.x + threadIdx.x;
  if (i < NEDGES) atomicAdd(&deg[(int)dst[i]], 1.0f);
}

__global__ void k_dinv(float* __restrict__ deg) {
  int i = blockIdx.x * blockDim.x + threadIdx.x;
  if (i < NNODES) deg[i] = rsqrtf(deg[i]);  // deg >= 1 always
}

// ---------------- dense GEMM via V_WMMA_F32_16X16X4_F32 ----------------
// H[M, 64] = act(X)[M, K] @ W[K, 64], act = PREACT ? relu(x + bias[k]) : x.
// Block = 128 threads = 4 waves; block owns 32 rows (two 16-row M-tiles,
// sharing B fragments), wave w owns output cols [16w, 16w+16).
template <int K, bool PREACT>
__global__ __launch_bounds__(128) void k_gemm_wmma(const float* __restrict__ X,
                                                   const float* __restrict__ W,
                                                   const float* __restrict__ bias,
                                                   float* __restrict__ H) {
  const int wave    = threadIdx.x >> 5;
  const int lane    = threadIdx.x & 31;
  const int rowBase = blockIdx.x * 32;
  const int row0    = rowBase + (lane & 15);     // A row, M-tile 0
  const int row1    = row0 + 16;                 // A row, M-tile 1
  const int col     = wave * 16 + (lane & 15);   // B/D column
  const int hiK     = (lane >> 4) << 1;          // lanes 0-15 -> K{0,1}; 16-31 -> K{2,3}

  v8f c0 = {};
  v8f c1 = {};
#pragma unroll
  for (int k0 = 0; k0 < K; k0 += 4) {
    // B fragment (4x16): shared by both M-tiles
    v2f b;
    b.x = W[(size_t)(k0 + hiK) * DHID + col];
    b.y = W[(size_t)(k0 + hiK + 1) * DHID + col];
    // A fragments: contiguous float2 along K
    v2f a0 = *(const v2f*)(X + (size_t)row0 * K + k0 + hiK);
    v2f a1 = *(const v2f*)(X + (size_t)row1 * K + k0 + hiK);
    if (PREACT) {  // fused bias+relu of the previous layer (bias indexes K dim)
      v2f bb = *(const v2f*)(bias + k0 + hiK);
      a0.x = fmaxf(a0.x + bb.x, 0.0f);  a0.y = fmaxf(a0.y + bb.y, 0.0f);
      a1.x = fmaxf(a1.x + bb.x, 0.0f);  a1.y = fmaxf(a1.y + bb.y, 0.0f);
    }
    c0 = __builtin_amdgcn_wmma_f32_16x16x4_f32(false, a0, false, b, (short)0, c0, false, false);
    c1 = __builtin_amdgcn_wmma_f32_16x16x4_f32(false, a1, false, b, (short)0, c1, false, false);
  }

  // D layout: VGPR r -> row tileBase + r + (lane<16 ? 0 : 8)
  const int rOff = (lane >> 4) << 3;
#pragma unroll
  for (int r = 0; r < 8; ++r) {
    H[(size_t)(rowBase + rOff + r) * DHID + col]      = c0[r];
    H[(size_t)(rowBase + 16 + rOff + r) * DHID + col] = c1[r];
  }
}

// ---------------- GCN aggregation ----------------
// out[i][c] = h[i][c] * dinv[i]^2   (self-loop contribution doubles as buffer init)
__global__ void k_self_init(const float* __restrict__ h, const float* __restrict__ dinv,
                            float* __restrict__ out) {
  int i = blockIdx.x * blockDim.x + threadIdx.x;   // over NNODES*16 float4's
  if (i < NNODES * (DHID / 4)) {
    float di = dinv[i >> 4];
    v4f hv = ((const v4f*)h)[i];
    v4f o;
    o.x = hv.x * di * di;  o.y = hv.y * di * di;
    o.z = hv.z * di * di;  o.w = hv.w * di * di;
    ((v4f*)out)[i] = o;
  }
}

// out[dst][c] += h[src][c] * dinv[src]*dinv[dst]; 16 threads/edge, float4 payload
__global__ void k_edge_agg(const long long* __restrict__ src, const long long* __restrict__ dst,
                           const float* __restrict__ h, const float* __restrict__ dinv,
                           float* __restrict__ out) {
  long long t = (long long)blockIdx.x * blockDim.x + threadIdx.x;
  int e  = (int)(t >> 4);
  int cb = (int)(t & 15) << 2;
  if (e < NEDGES) {
    int s = (int)src[e];
    int d = (int)dst[e];
    float norm = dinv[s] * dinv[d];
    v4f hv = *(const v4f*)(h + (size_t)s * DHID + cb);
    float* o = out + (size_t)d * DHID + cb;
    atomicAdd(o + 0, hv.x * norm);
    atomicAdd(o + 1, hv.y * norm);
    atomicAdd(o + 2, hv.z * norm);
    atomicAdd(o + 3, hv.w * norm);
  }
}

// ---------------- pooling + classifier ----------------
__global__ void k_zero(float* __restrict__ p, int n) {
  int i = blockIdx.x * blockDim.x + threadIdx.x;
  if (i < n) p[i] = 0.0f;
}

__global__ void k_pool_sum(const float* __restrict__ h, const long long* __restrict__ batch,
                           float* __restrict__ sums) {
  int i = blockIdx.x * blockDim.x + threadIdx.x;   // over NNODES*16 float4's
  if (i < NNODES * (DHID / 4)) {
    int node = i >> 4;
    int cb = (i & 15) << 2;
    v4f hv = ((const v4f*)h)[i];
    float* o = sums + (size_t)batch[node] * DHID + cb;
    atomicAdd(o + 0, hv.x);
    atomicAdd(o + 1, hv.y);
    atomicAdd(o + 2, hv.z);
    atomicAdd(o + 3, hv.w);
  }
}

__global__ void k_pool_cnt(const long long* __restrict__ batch, float* __restrict__ cnt) {
  int i = blockIdx.x * blockDim.x + threadIdx.x;
  if (i < NNODES) atomicAdd(&cnt[(int)batch[i]], 1.0f);
}

// out[g][j] = bl[j] + sum_c (sums[g][c]/max(cnt,1) + b3[c]) * Wl[c][j]
// (layer-3 bias folded in: mean(h + b3) == mean(h) + b3)
__global__ void k_final(const float* __restrict__ sums, const float* __restrict__ cnt,
                        const float* __restrict__ b3, const float* __restrict__ Wl,
                        const float* __restrict__ bl, float* __restrict__ out) {
  int i = blockIdx.x * blockDim.x + threadIdx.x;
  if (i < NGRAPHS * NCLS) {
    int g = i / NCLS, j = i % NCLS;
    float inv = 1.0f / fmaxf(cnt[g], 1.0f);
    float acc = bl[j];
#pragma unroll
    for (int c = 0; c < DHID; ++c)
      acc += (sums[(size_t)g * DHID + c] * inv + b3[c]) * Wl[c * NCLS + j];
    out[i] = acc;
  }
}

// ---------------- driver ----------------
extern "C" void kernel_launch(void* const* d_in, const int* in_sizes, int n_in,
                              void* d_out, int out_size, void* d_ws, size_t ws_size,
                              hipStream_t stream) {
  (void)in_sizes; (void)n_in; (void)out_size; (void)ws_size;
  const float*     x     = (const float*)d_in[0];
  const long long* ei    = (const long long*)d_in[1];   // int64 [2, E]
  const long long* src   = ei;
  const long long* dstv  = ei + NEDGES;
  const long long* batch = (const long long*)d_in[2];   // int64 [N]
  const float* W1 = (const float*)d_in[3];  const float* b1 = (const float*)d_in[4];
  const float* W2 = (const float*)d_in[5];  const float* b2 = (const float*)d_in[6];
  const float* W3 = (const float*)d_in[7];  const float* b3 = (const float*)d_in[8];
  const float* Wl = (const float*)d_in[9];  const float* bl = (const float*)d_in[10];
  float* out = (float*)d_out;

  // workspace carve-up (floats): dinv | hA | hB | pool | cnt  (~52 MB)
  float* ws   = (float*)d_ws;
  float* dinv = ws;
  float* hA   = dinv + ((NNODES + 255) & ~255);
  float* hB   = hA + (size_t)NNODES * DHID;
  float* pool = hB + (size_t)NNODES * DHID;
  float* cnt  = pool + NGRAPHS * DHID;

  const int B = 256;
  const int gNodes  = (NNODES + B - 1) / B;
  const int gEdges  = (NEDGES + B - 1) / B;
  const int gNode4  = (NNODES * (DHID / 4) + B - 1) / B;                   // float4 grids
  const int gEdge4  = (int)(((long long)NEDGES * (DHID / 4) + B - 1) / B);
  const int gGemm   = NNODES / 32;  // 100000 % 32 == 0

  // symmetric normalization (computed once, reused by all 3 layers)
  k_deg_init<<<gNodes, B, 0, stream>>>(dinv);
  k_deg_edges<<<gEdges, B, 0, stream>>>(dstv, dinv);
  k_dinv<<<gNodes, B, 0, stream>>>(dinv);

  // layer 1: hA = x @ W1 ; hB = pre-bias aggregated output
  k_gemm_wmma<DIN, false><<<gGemm, 128, 0, stream>>>(x, W1, nullptr, hA);
  k_self_init<<<gNode4, B, 0, stream>>>(hA, dinv, hB);
  k_edge_agg<<<gEdge4, B, 0, stream>>>(src, dstv, hA, dinv, hB);

  // layer 2: relu(hB + b1) fused into GEMM A-load
  k_gemm_wmma<DHID, true><<<gGemm, 128, 0, stream>>>(hB, W2, b1, hA);
  k_self_init<<<gNode4, B, 0, stream>>>(hA, dinv, hB);
  k_edge_agg<<<gEdge4, B, 0, stream>>>(src, dstv, hA, dinv, hB);

  // layer 3: relu(hB + b2) fused into GEMM A-load; b3 folded into classifier
  k_gemm_wmma<DHID, true><<<gGemm, 128, 0, stream>>>(hB, W3, b2, hA);
  k_self_init<<<gNode4, B, 0, stream>>>(hA, dinv, hB);
  k_edge_agg<<<gEdge4, B, 0, stream>>>(src, dstv, hA, dinv, hB);

  // global mean pool + classifier
  k_zero<<<(NGRAPHS * DHID + NGRAPHS + B - 1) / B, B, 0, stream>>>(pool, NGRAPHS * DHID + NGRAPHS);
  k_pool_sum<<<gNode4, B, 0, stream>>>(hB, batch, pool);
  k_pool_cnt<<<gNodes, B, 0, stream>>>(batch, cnt);
  k_final<<<(NGRAPHS * NCLS + B - 1) / B, B, 0, stream>>>(pool, cnt, b3, Wl, bl, out);
}